// LocalFeatureBuilder_90117003804920
// MI455X (gfx1250) — compile-verified
//
#include <hip/hip_runtime.h>
#include <stdint.h>

// ---------------------------------------------------------------------------
// LocalFeatureBuilder for MI455X (gfx1250, wave32).
//  B=4, N=4096, Q=4096, K=32, EMB=32, RBF=16, FEAT=3+1+32+16+1=53
//  dist^2(q,a) = q.q - 2 q.a + a.a  ->  V_WMMA_F32_16X16X4_F32 tile GEMM
// ---------------------------------------------------------------------------

namespace {
constexpr int   BATCH = 4;
constexpr int   NATOM = 4096;
constexpr int   NQ    = 4096;
constexpr int   KNB   = 32;     // neighbors == wave32 width
constexpr int   EMB   = 32;
constexpr int   NRBF  = 16;
constexpr int   FEAT  = 53;
constexpr float CUT   = 5.0f;
constexpr float GAMMA = 10.24f; // 1/(CUT/NRBF)^2
constexpr float EPSF  = 1e-12f;

constexpr int CHUNK     = 1024;         // atoms per LDS distance chunk
constexpr int DSTRIDE   = 1028;         // f32 row stride (bank-conflict free halves)
constexpr int NWAVES    = 16;           // one wave per query of the 16-query tile
constexpr int BLOCK     = NWAVES * 32;
constexpr int SLICE_U64 = 1024;         // 8KB per-wave scratch (32 lanes x 32 u64)
}

typedef __attribute__((ext_vector_type(2))) float v2f;
typedef __attribute__((ext_vector_type(8))) float v8f;

__global__ __launch_bounds__(BLOCK)
void lfb_kernel(const float* __restrict__ coords,       // [B,N,3]
                const int*   __restrict__ atom_types,   // [B,N]
                const float* __restrict__ radii,        // [B,N]
                const float* __restrict__ qpts,         // [B,Q,3]
                const unsigned char* __restrict__ amask,// [B,N] bool
                const unsigned char* __restrict__ qmask,// [B,Q] bool
                const float* __restrict__ etab,         // [100,EMB]
                const float* __restrict__ rbfc,         // [NRBF]
                float* __restrict__ out_feat,           // [B,Q,K,FEAT]
                float* __restrict__ out_mask,           // [B,Q,K]
                float* __restrict__ out_idx,            // [B,Q,K]
                float* __restrict__ out_dist)           // [B,Q,K]
{
  // 131072 B union region: distance tiles, then merge scratch, then staging.
  __shared__ unsigned long long sh_u64[NWAVES * SLICE_U64];
  __shared__ float sh_embed[100 * EMB];   // 12.8 KB embedding cache
  __shared__ float sh_qn[16];
  __shared__ float sh_rbf[NRBF];

  float* sh_dist = reinterpret_cast<float*>(sh_u64);

  const int tid  = threadIdx.x;
  const int lane = tid & 31;
  const int wave = tid >> 5;
  const int b    = blockIdx.x >> 8;          // NQ/16 == 256 query tiles
  const int q0   = (blockIdx.x & 255) << 4;

  for (int i = tid; i < 100 * EMB; i += BLOCK) sh_embed[i] = etab[i];
  if (tid < NRBF) sh_rbf[tid] = rbfc[tid];

  // ---- A matrix (16 queries x [qx,qy,qz,1]) + per-row q.q --------------
  const int m = lane & 15;
  const float qxm = qpts[(size_t)(b * NQ + q0 + m) * 3 + 0];
  const float qym = qpts[(size_t)(b * NQ + q0 + m) * 3 + 1];
  const float qzm = qpts[(size_t)(b * NQ + q0 + m) * 3 + 2];
  if (wave == 0 && lane < 16) sh_qn[lane] = qxm * qxm + qym * qym + qzm * qzm;
  v2f Amat;                                   // f32 A 16x4: lanes0-15 K=0,1; lanes16-31 K=2,3
  Amat.x = (lane < 16) ? qxm : qzm;
  Amat.y = (lane < 16) ? qym : 1.0f;
  __syncthreads();

  const int mhalf = (lane >> 4) << 3;         // C-row offset: lanes16-31 hold M=v+8
  float qn8[8];
  #pragma unroll
  for (int v = 0; v < 8; ++v) qn8[v] = sh_qn[v + mhalf];

  // ---- per-lane sorted top-K list of (d2bits<<32 | atom index) ---------
  unsigned long long lst[KNB];
  #pragma unroll
  for (int i = 0; i < KNB; ++i) lst[i] = ~0ull;

  for (int cbase = 0; cbase < NATOM; cbase += CHUNK) {
    // -- distance phase: WMMA 16x16 tiles, 4 tiles per wave per chunk ----
    for (int t = wave; t < CHUNK / 16; t += NWAVES) {
      const int aidx = cbase + t * 16 + m;    // B column n == m
      const float ax = coords[(size_t)(b * NATOM + aidx) * 3 + 0];
      const float ay = coords[(size_t)(b * NATOM + aidx) * 3 + 1];
      const float az = coords[(size_t)(b * NATOM + aidx) * 3 + 2];
      v2f Bmat;                               // rows: -2ax,-2ay | -2az, a.a
      Bmat.x = (lane < 16) ? (-2.0f * ax) : (-2.0f * az);
      Bmat.y = (lane < 16) ? (-2.0f * ay) : (ax * ax + ay * ay + az * az);
      v8f C;
      #pragma unroll
      for (int v = 0; v < 8; ++v) C[v] = qn8[v];   // seed with q.q
      C = __builtin_amdgcn_wmma_f32_16x16x4_f32(false, Amat, false, Bmat,
                                                (short)0, C, false, false);
      const float inj =
          (amask[(size_t)b * NATOM + aidx] != 0) ? 0.0f : __builtin_inff();
      #pragma unroll
      for (int v = 0; v < 8; ++v)
        sh_dist[(v + mhalf) * DSTRIDE + t * 16 + m] = C[v] + inj;
    }
    __syncthreads();

    // -- selection phase: wave scans its query row, streaming insert ----
    #pragma unroll 1
    for (int j = lane; j < CHUNK; j += 32) {
      const float d2 = sh_dist[wave * DSTRIDE + j];
      const unsigned long long p =
          ((unsigned long long)__float_as_uint(d2) << 32) |
          (unsigned int)(cbase + j);
      if (p < lst[KNB - 1]) {
        unsigned long long cur = p;
        #pragma unroll
        for (int i = 0; i < KNB; ++i) {
          const unsigned long long lo = (cur < lst[i]) ? cur : lst[i];
          cur = (cur < lst[i]) ? lst[i] : cur;
          lst[i] = lo;
        }
      }
    }
    __syncthreads();
  }

  // ---- merge 32 sorted per-lane lists -> lane j holds j-th smallest ----
  unsigned long long* slice = sh_u64 + (size_t)wave * SLICE_U64;
  #pragma unroll
  for (int i = 0; i < KNB; ++i) slice[lane * KNB + i] = lst[i];
  __syncthreads();

  int ptr = 1;
  unsigned long long head = slice[lane * KNB];
  unsigned long long mine = ~0ull;
  #pragma unroll 1
  for (int j = 0; j < KNB; ++j) {
    unsigned long long mn = head;
    #pragma unroll
    for (int off = 16; off > 0; off >>= 1) {
      const unsigned long long o = __shfl_xor(mn, off, 32);
      mn = (o < mn) ? o : mn;
    }
    if (lane == j) mine = mn;
    if (head == mn) {                         // packed keys are unique
      head = (ptr < KNB) ? slice[lane * KNB + ptr] : ~0ull;
      ++ptr;
    }
  }
  __syncthreads();

  // ---- feature phase: wave = query (q0+wave), lane = neighbor ----------
  const int q = q0 + wave;
  const bool qvalid = qmask[(size_t)b * NQ + q] != 0;
  const float qx = qpts[(size_t)(b * NQ + q) * 3 + 0];
  const float qy = qpts[(size_t)(b * NQ + q) * 3 + 1];
  const float qz = qpts[(size_t)(b * NQ + q) * 3 + 2];

  const int idx = (int)(unsigned int)(mine & 0xFFFFFFFFull);
  const bool avalid = amask[(size_t)b * NATOM + idx] != 0;
  const float ax = coords[(size_t)(b * NATOM + idx) * 3 + 0];
  const float ay = coords[(size_t)(b * NATOM + idx) * 3 + 1];
  const float az = coords[(size_t)(b * NATOM + idx) * 3 + 2];
  const float dx = qx - ax, dy = qy - ay, dz = qz - az;
  // exact reference distance (not the WMMA expansion)
  const float d = sqrtf(fmaf(dx, dx, fmaf(dy, dy, fmaf(dz, dz, EPSF))));
  const bool nm = avalid && (d <= CUT) && qvalid;
  const float z = nm ? 1.0f : 0.0f;

  float* stg = reinterpret_cast<float*>(slice);   // reuse: 32*53 f32 staging
  float* fv  = stg + lane * FEAT;
  fv[0] = dx * z;  fv[1] = dy * z;  fv[2] = dz * z;
  fv[3] = radii[(size_t)b * NATOM + idx] * z;
  const int ty = atom_types[(size_t)b * NATOM + idx];
  #pragma unroll
  for (int e = 0; e < EMB; ++e) fv[4 + e] = sh_embed[ty * EMB + e] * z;
  #pragma unroll
  for (int r = 0; r < NRBF; ++r) {
    const float dd = d - sh_rbf[r];
    fv[36 + r] = __expf(-GAMMA * dd * dd) * z;
  }
  fv[52] = d * z;
  __syncthreads();

  const size_t fb = (size_t)(b * NQ + q) * KNB * FEAT;
  for (int t = lane; t < KNB * FEAT; t += 32)     // coalesced feature stream
    out_feat[fb + t] = stg[t];

  const size_t sb = (size_t)(b * NQ + q) * KNB + lane;
  out_mask[sb] = nm ? 1.0f : 0.0f;
  out_idx[sb]  = avalid ? (float)idx : -1.0f;
  out_dist[sb] = nm ? d : 0.0f;
}

extern "C" void kernel_launch(void* const* d_in, const int* in_sizes, int n_in,
                              void* d_out, int out_size, void* d_ws, size_t ws_size,
                              hipStream_t stream) {
  (void)in_sizes; (void)n_in; (void)out_size; (void)d_ws; (void)ws_size;
  const float*         coords     = (const float*)d_in[0];
  const int*           atom_types = (const int*)d_in[1];
  const float*         radii      = (const float*)d_in[2];
  const float*         qpts       = (const float*)d_in[3];
  const unsigned char* amask      = (const unsigned char*)d_in[4];
  const unsigned char* qmask      = (const unsigned char*)d_in[5];
  const float*         etab       = (const float*)d_in[6];
  const float*         rbfc       = (const float*)d_in[7];

  float* out_feat = (float*)d_out;
  float* out_mask = out_feat + (size_t)BATCH * NQ * KNB * FEAT;
  float* out_idx  = out_mask + (size_t)BATCH * NQ * KNB;
  float* out_dst  = out_idx  + (size_t)BATCH * NQ * KNB;

  dim3 grid(BATCH * (NQ / 16));   // one workgroup per 16-query tile
  lfb_kernel<<<grid, BLOCK, 0, stream>>>(coords, atom_types, radii, qpts,
                                         amask, qmask, etab, rbfc,
                                         out_feat, out_mask, out_idx, out_dst);
}